// TreeSelfAttentionHybrid_30116310680240
// MI455X (gfx1250) — compile-verified
//
#include <hip/hip_runtime.h>
#include <hip/hip_bf16.h>

typedef __attribute__((ext_vector_type(2))) float v2f;
typedef __attribute__((ext_vector_type(8))) float v8f;

#define D_DIM 1024
#define B_DIM 8
#define T_DIM 4096
#define NODE_STRIDE (16 * D_DIM)   // floats per node: 16 rows (8 batch + 8 zero-pad)

// ---------------------------------------------------------------------------
// Kernel 1: pooling into zero-padded node buffers.
//   leaves[n][r][d] = mean_t x[b=r, n*256 + t, d]  for r < 8
//                   = 0                             for r in 8..15 (WMMA pad rows)
// ---------------------------------------------------------------------------
__global__ void pool_kernel(const float* __restrict__ x, float* __restrict__ leaves) {
    int idx = blockIdx.x * blockDim.x + threadIdx.x;   // 8 * 16 * 1024 = 131072
    int d  = idx & (D_DIM - 1);
    int r  = (idx >> 10) & 15;
    int n  = idx >> 14;
    float val = 0.0f;
    if (r < B_DIM) {
        const float* xp = x + ((size_t)r * T_DIM + (size_t)n * 256) * D_DIM + d;
        float acc = 0.0f;
#pragma unroll 8
        for (int t = 0; t < 256; ++t) acc += xp[(size_t)t * D_DIM];
        val = acc * (1.0f / 256.0f);
    }
    leaves[idx] = val;
}

// ---------------------------------------------------------------------------
// Kernel 2: one tree level.  For node j of this level:
//   fused(16 x 2048) = [Sin[c0(j)], Sin[c1(j)]]   (rows 8..15 are zero padding;
//                                                  c0==c1==j at leaf level)
//   proj = fused @ W[start+j] + bias[start+j]
//   Sout[j] = proj * (proj - bt[start+j] > 0), rows 8..15 stored as zero
// One wave per 16x16 output tile via V_WMMA_F32_16X16X4_F32 (exact f32 path).
// ---------------------------------------------------------------------------
__global__ __launch_bounds__(32)
void tree_level_kernel(const float* __restrict__ Sin, float* __restrict__ Sout,
                       const float* __restrict__ W, const float* __restrict__ bias,
                       const float* __restrict__ bt, int startNode, int leafMode) {
    const int j    = blockIdx.y;              // node within level
    const int n0   = blockIdx.x * 16;         // output column tile
    const int lane = threadIdx.x;             // 0..31, wave32
    const int half = lane >> 4;               // 0 or 1 (K-offset selector)
    const int lm   = lane & 15;

    const int node = startNode + j;
    const int c0 = leafMode ? j : (2 * j);
    const int c1 = leafMode ? j : (2 * j + 1);

    // A-matrix (16x4 f32): M = lane&15; VGPR0 holds K = half*2, VGPR1 K = half*2+1.
    // Rows 8..15 of Sin are pre-zeroed, so loads need no masking.
    const int M = lm;
    // B-matrix (4x16 f32): N = n0 + (lane&15), rows K = half*2, half*2+1
    const int N = n0 + lm;

    const float* A0 = Sin + (size_t)c0 * NODE_STRIDE + (size_t)M * D_DIM; // k in [0,1024)
    const float* A1 = Sin + (size_t)c1 * NODE_STRIDE + (size_t)M * D_DIM; // k-1024
    const float* Wn = W + (size_t)node * (2 * D_DIM) * D_DIM;             // 2048 x 1024

    v8f c = {0.f, 0.f, 0.f, 0.f, 0.f, 0.f, 0.f, 0.f};

    // K halves: [0,1024) reads child c0, [1024,2048) reads child c1.
#pragma unroll 1
    for (int k0 = 0; k0 < D_DIM; k0 += 32) {
        __builtin_prefetch(Wn + (size_t)(k0 + 64) * D_DIM + N, 0, 3);
#pragma unroll
        for (int kk = 0; kk < 32; kk += 4) {
            const int ka = k0 + kk + half * 2;
            v2f a = *(const v2f*)(A0 + ka);
            v2f bf;
            const float* wp = Wn + (size_t)ka * D_DIM + N;
            bf.x = wp[0];
            bf.y = wp[D_DIM];
            c = __builtin_amdgcn_wmma_f32_16x16x4_f32(false, a, false, bf,
                                                      (short)0, c, false, false);
        }
    }
#pragma unroll 1
    for (int k0 = 0; k0 < D_DIM; k0 += 32) {
        __builtin_prefetch(Wn + (size_t)(D_DIM + k0 + 64) * D_DIM + N, 0, 3);
#pragma unroll
        for (int kk = 0; kk < 32; kk += 4) {
            const int ka = k0 + kk + half * 2;
            v2f a = *(const v2f*)(A1 + ka);
            v2f bf;
            const float* wp = Wn + (size_t)(D_DIM + ka) * D_DIM + N;
            bf.x = wp[0];
            bf.y = wp[D_DIM];
            c = __builtin_amdgcn_wmma_f32_16x16x4_f32(false, a, false, bf,
                                                      (short)0, c, false, false);
        }
    }

    // Epilogue: each lane owns column N; VGPR v -> row M = v + half*8.
    // Rows 0..7 (half==0) get gated states; rows 8..15 (half==1) keep the
    // zero padding invariant for the next level's A loads.
    const float bb = bias[(size_t)node * D_DIM + N];
    const float tt = bt[(size_t)node * D_DIM + N];
    float* outp = Sout + (size_t)j * NODE_STRIDE + (size_t)(half * 8) * D_DIM + N;
#pragma unroll
    for (int v = 0; v < 8; ++v) {
        float p = c[v] + bb;
        float s = (p - tt > 0.0f) ? p : 0.0f;   // proj * hard-step(proj - bt)
        outp[(size_t)v * D_DIM] = (half == 0) ? s : 0.0f;
    }
}

// ---------------------------------------------------------------------------
// Kernel 3: out[b,t,:] = layernorm(root[b,:] + x[b,t,:]) * ln_w + ln_b
// One 256-thread block per row; 4 elements per thread; LDS reduction.
// ---------------------------------------------------------------------------
__global__ __launch_bounds__(256)
void ln_residual_kernel(const float* __restrict__ x, const float* __restrict__ root,
                        const float* __restrict__ ln_w, const float* __restrict__ ln_b,
                        float* __restrict__ out) {
    const int row = blockIdx.x;            // 0 .. B*T-1
    const int b   = row >> 12;             // row / 4096
    const int tid = threadIdx.x;
    const float* xr = x + (size_t)row * D_DIM;
    const float* rb = root + (size_t)b * D_DIM;

    float v[4];
    float sum = 0.f, sq = 0.f;
#pragma unroll
    for (int i = 0; i < 4; ++i) {
        const int d = tid + i * 256;
        const float t = xr[d] + rb[d];
        v[i] = t;
        sum += t;
        sq  += t * t;
    }

    __shared__ float s1[256];
    __shared__ float s2[256];
    s1[tid] = sum; s2[tid] = sq;
    __syncthreads();
#pragma unroll
    for (int off = 128; off > 0; off >>= 1) {
        if (tid < off) { s1[tid] += s1[tid + off]; s2[tid] += s2[tid + off]; }
        __syncthreads();
    }
    const float mean = s1[0] * (1.0f / D_DIM);
    const float var  = s2[0] * (1.0f / D_DIM) - mean * mean;
    const float rstd = __frsqrt_rn(var + 1e-5f);

    float* o = out + (size_t)row * D_DIM;
#pragma unroll
    for (int i = 0; i < 4; ++i) {
        const int d = tid + i * 256;
        o[d] = (v[i] - mean) * rstd * ln_w[d] + ln_b[d];
    }
}

// ---------------------------------------------------------------------------
// Launch
// ---------------------------------------------------------------------------
extern "C" void kernel_launch(void* const* d_in, const int* in_sizes, int n_in,
                              void* d_out, int out_size, void* d_ws, size_t ws_size,
                              hipStream_t stream) {
    const float* x    = (const float*)d_in[0];  // (8, 4096, 1024)
    const float* W    = (const float*)d_in[1];  // (15, 2048, 1024)
    const float* bias = (const float*)d_in[2];  // (15, 1024)
    const float* bt   = (const float*)d_in[3];  // (15, 1024)
    const float* ln_w = (const float*)d_in[4];  // (1024,)
    const float* ln_b = (const float*)d_in[5];  // (1024,)
    float* out = (float*)d_out;

    float* ws0 = (float*)d_ws;                  // leaves  : 8 nodes * 16 * 1024
    float* ws1 = ws0 + 8 * NODE_STRIDE;         // states A
    float* ws2 = ws1 + 8 * NODE_STRIDE;         // states B

    // 1. pool first 2048 timesteps into 8 zero-padded leaves
    pool_kernel<<<(8 * NODE_STRIDE) / 256, 256, 0, stream>>>(x, ws0);

    // 2. tree levels (dependent): 8 -> 4 -> 2 -> 1 nodes
    tree_level_kernel<<<dim3(D_DIM / 16, 8), 32, 0, stream>>>(ws0, ws1, W, bias, bt, 7, 1);
    tree_level_kernel<<<dim3(D_DIM / 16, 4), 32, 0, stream>>>(ws1, ws2, W, bias, bt, 3, 0);
    tree_level_kernel<<<dim3(D_DIM / 16, 2), 32, 0, stream>>>(ws2, ws1, W, bias, bt, 1, 0);
    tree_level_kernel<<<dim3(D_DIM / 16, 1), 32, 0, stream>>>(ws1, ws2, W, bias, bt, 0, 0);

    // 3. residual + layernorm over full x
    ln_residual_kernel<<<B_DIM * T_DIM, 256, 0, stream>>>(x, ws2, ln_w, ln_b, out);
}